// GNNModel_86131274154307
// MI455X (gfx1250) — compile-verified
//
#include <hip/hip_runtime.h>
#include <math.h>

// ---------------- problem dims (fixed by reference) ----------------
#define NN   100000   // nodes  (multiple of 16 and 8)
#define NE   800000   // edges  (multiple of 16 and 8)
#define FEAT 128
#define HC   64       // heads*channels
#define EDIM 16
#define EMB  32
#define NCLS 16
#define NLAY 2
#define BN_EPS 1e-5f
#define ATT_SCALE 0.17677669529663687f  // 1/sqrt(32)

typedef __attribute__((ext_vector_type(2))) float v2f;
typedef __attribute__((ext_vector_type(8))) float v8f;
typedef __attribute__((ext_vector_type(4))) unsigned int v4u;
typedef __attribute__((ext_vector_type(8))) int v8i_;
typedef __attribute__((ext_vector_type(4))) int v4i_;

// ---- fp32 WMMA: D(16x16) = A(16x4) * B(4x16) + C, exact fp32 path ----
__device__ __forceinline__ v8f wmma_f32x4(v2f a, v2f b, v8f c) {
  return __builtin_amdgcn_wmma_f32_16x16x4_f32(
      /*neg_a=*/false, a, /*neg_b=*/false, b,
      /*c_mod=*/(short)0, c, /*reuse_a=*/false, /*reuse_b=*/false);
}

// ---- TDM: DMA a row-major f32 matrix [rows, cols] into LDS offset 0 ----
// D# per CDNA5 ISA ch.8: group0 = {count, lds_addr, global_addr, type=2},
// group1 = {data_size=4B, tensor dims, tile dims, dim0 stride}, groups 2/3 zero (2-D).
// This toolchain exposes the 6-arg builtin: (v4u, v8i, v4i, v4i, v8i, i32 cpol).
__device__ __forceinline__ void tdm_load_2d_f32(const float* W, int rows, int cols) {
  unsigned long long ga = (unsigned long long)W;
  v4u g0;
  g0[0] = 1u;                                               // count=1, no gather
  g0[1] = 0u;                                               // lds_addr = 0 (dynamic LDS base)
  g0[2] = (unsigned)(ga & 0xFFFFFFFFull);                   // global_addr[31:0]
  g0[3] = (unsigned)((ga >> 32) & 0x1FFFFFFull)             // global_addr[56:32]
        | (2u << 30);                                       // type = 2 ("image")
  v8i_ g1;
  g1[0] = (2 << 16);                                        // data_size=2 -> 4 bytes; wg_mask=0
  g1[1] = (cols & 0xFFFF) << 16;                            // tensor_dim0[15:0]
  g1[2] = ((cols >> 16) & 0xFFFF) | ((rows & 0xFFFF) << 16);// tensor_dim0[31:16] | tensor_dim1[15:0]
  g1[3] = ((rows >> 16) & 0xFFFF) | ((cols & 0xFFFF) << 16);// tensor_dim1[31:16] | tile_dim0
  g1[4] = (rows & 0xFFFF);                                  // tile_dim1 (tile_dim2 = 0)
  g1[5] = cols;                                             // tensor_dim0_stride[31:0]
  g1[6] = 0;                                                // stride[47:32] | dim1_stride[15:0]
  g1[7] = 0;
  v4i_ z4 = {0, 0, 0, 0};
  v8i_ z8 = {0, 0, 0, 0, 0, 0, 0, 0};
  __builtin_amdgcn_tensor_load_to_lds(g0, g1, z4, z4, z8, 0);
}

// ---------------- utility ----------------
__global__ void fill_kernel(float* __restrict__ p, float v, int n) {
  int i = blockIdx.x * blockDim.x + threadIdx.x;
  if (i < n) p[i] = v;
}

__device__ __forceinline__ void atomicMaxF(float* addr, float val) {
  if (val >= 0.f) atomicMax((int*)addr, __float_as_int(val));
  else            atomicMin((unsigned int*)addr, __float_as_uint(val));
}

__device__ __forceinline__ float wave_sum(float v) {
  #pragma unroll
  for (int off = 16; off > 0; off >>= 1) v += __shfl_xor(v, off, 32);
  return v;
}

// ---------------- fused 4-projection GEMM (Q,K,V,skip) ----------------
// block = 128 threads = 4 waves; wave w computes its matrix for 16 rows.
// A frag (16x4 f32): lanes 0-15 -> M=lane, K={0,1}; lanes 16-31 -> K={2,3}.
// C/D (16x16 f32):   vgpr r: lanes 0-15 -> M=r, lanes 16-31 -> M=r+8, N=lane&15.
__global__ void proj4_kernel(const float* __restrict__ X, int Fin,
    const float* __restrict__ Wq, const float* __restrict__ bq,
    const float* __restrict__ Wk, const float* __restrict__ bk,
    const float* __restrict__ Wv, const float* __restrict__ bv,
    const float* __restrict__ Ws, const float* __restrict__ bs,
    float* __restrict__ Q, float* __restrict__ K2,
    float* __restrict__ V2, float* __restrict__ S2)
{
  const int wave = threadIdx.x >> 5;
  const int lane = threadIdx.x & 31;
  const int row0 = blockIdx.x << 4;
  const float *W, *B; float* O;
  switch (wave) {
    case 0:  W = Wq; B = bq; O = Q;  break;
    case 1:  W = Wk; B = bk; O = K2; break;
    case 2:  W = Wv; B = bv; O = V2; break;
    default: W = Ws; B = bs; O = S2; break;
  }
  const int m  = lane & 15;
  const int kh = (lane >> 4) << 1;           // 0 or 2
  v8f acc[4];
  #pragma unroll
  for (int t = 0; t < 4; ++t) {
    float bb = B[t * 16 + m];
    #pragma unroll
    for (int r = 0; r < 8; ++r) acc[t][r] = bb;
  }
  const float* xrow = X + (size_t)(row0 + m) * Fin;
  for (int kk = 0; kk < Fin; kk += 4) {
    v2f a; a.x = xrow[kk + kh]; a.y = xrow[kk + kh + 1];
    #pragma unroll
    for (int t = 0; t < 4; ++t) {
      v2f b;
      b.x = W[(size_t)(kk + kh)     * HC + t * 16 + m];
      b.y = W[(size_t)(kk + kh + 1) * HC + t * 16 + m];
      acc[t] = wmma_f32x4(a, b, acc[t]);
    }
  }
  const int rbase = row0 + ((lane >> 4) << 3);
  #pragma unroll
  for (int t = 0; t < 4; ++t)
    #pragma unroll
    for (int r = 0; r < 8; ++r)
      O[(size_t)(rbase + r) * HC + t * 16 + m] = acc[t][r];
}

// ---------------- generic single-matrix WMMA GEMM with TDM-staged B ------
// one wave per block; W[Kdim, NT*16] DMA'd into LDS by the Tensor Data Mover,
// B fragments then come from LDS while A streams from global.
template <int NT, bool RELU>
__global__ void gemm_kernel(const float* __restrict__ A, int Kdim,
                            const float* __restrict__ W,
                            const float* __restrict__ bias,
                            float* __restrict__ O)
{
  extern __shared__ float smemW[];
  const int ldb = NT * 16;
  tdm_load_2d_f32(W, Kdim, ldb);            // async: TENSORcnt++

  const int lane = threadIdx.x & 31;
  const int row0 = blockIdx.x << 4;
  const int m  = lane & 15;
  const int kh = (lane >> 4) << 1;
  v8f acc[NT];
  #pragma unroll
  for (int t = 0; t < NT; ++t) {
    float bb = bias ? bias[t * 16 + m] : 0.0f;
    #pragma unroll
    for (int r = 0; r < 8; ++r) acc[t][r] = bb;
  }
  const float* arow = A + (size_t)(row0 + m) * Kdim;
  __builtin_amdgcn_s_wait_tensorcnt(0);     // LDS tile ready
  for (int kk = 0; kk < Kdim; kk += 4) {
    v2f a; a.x = arow[kk + kh]; a.y = arow[kk + kh + 1];
    #pragma unroll
    for (int t = 0; t < NT; ++t) {
      v2f b;
      b.x = smemW[(kk + kh)     * ldb + t * 16 + m];
      b.y = smemW[(kk + kh + 1) * ldb + t * 16 + m];
      acc[t] = wmma_f32x4(a, b, acc[t]);
    }
  }
  const int rbase = row0 + ((lane >> 4) << 3);
  #pragma unroll
  for (int t = 0; t < NT; ++t)
    #pragma unroll
    for (int r = 0; r < 8; ++r) {
      float v = acc[t][r];
      if (RELU) v = v > 0.f ? v : 0.f;
      O[(size_t)(rbase + r) * ldb + t * 16 + m] = v;
    }
}

// ---------------- edge phase: wave-per-edge, coalesced ----------------
// lane k owns channels k (head 0) and k+32 (head 1).
__global__ void edge_alpha_kernel(const int* __restrict__ src, const int* __restrict__ dst,
    const float* __restrict__ Q, const float* __restrict__ Kf,
    const float* __restrict__ EP, float* __restrict__ ALPHA, float* __restrict__ Mx)
{
  const int lane = threadIdx.x & 31;
  const int e = blockIdx.x * (blockDim.x >> 5) + (threadIdx.x >> 5);
  const int s = src[e], d = dst[e];
  const float* q  = Q  + (size_t)d * HC;
  const float* k  = Kf + (size_t)s * HC;
  const float* ep = EP + (size_t)e * HC;
  float p0 = q[lane]      * (k[lane]      + ep[lane]);
  float p1 = q[lane + 32] * (k[lane + 32] + ep[lane + 32]);
  float a0 = wave_sum(p0) * ATT_SCALE;
  float a1 = wave_sum(p1) * ATT_SCALE;
  if (lane == 0) {
    ALPHA[2 * (size_t)e]     = a0;
    ALPHA[2 * (size_t)e + 1] = a1;
    atomicMaxF(&Mx[2 * (size_t)d],     a0);
    atomicMaxF(&Mx[2 * (size_t)d + 1], a1);
  }
}

// fused: exp(alpha - max), segment-sum of weights, weighted V aggregation
__global__ void edge_agg_kernel(const int* __restrict__ src, const int* __restrict__ dst,
    const float* __restrict__ Vf, const float* __restrict__ EP,
    const float* __restrict__ ALPHA, const float* __restrict__ Mx,
    float* __restrict__ Sm, float* __restrict__ AGG)
{
  const int lane = threadIdx.x & 31;
  const int e = blockIdx.x * (blockDim.x >> 5) + (threadIdx.x >> 5);
  const int s = src[e], d = dst[e];
  const float a0 = __expf(ALPHA[2 * (size_t)e]     - Mx[2 * (size_t)d]);
  const float a1 = __expf(ALPHA[2 * (size_t)e + 1] - Mx[2 * (size_t)d + 1]);
  if (lane == 0) {
    atomicAdd(&Sm[2 * (size_t)d],     a0);
    atomicAdd(&Sm[2 * (size_t)d + 1], a1);
  }
  const float* v  = Vf + (size_t)s * HC;
  const float* ep = EP + (size_t)e * HC;
  float* o = AGG + (size_t)d * HC;
  atomicAdd(&o[lane],      (v[lane]      + ep[lane])      * a0);
  atomicAdd(&o[lane + 32], (v[lane + 32] + ep[lane + 32]) * a1);
}

// wave-per-node: softmax-normalize + beta-gated skip (in place on AGG)
__global__ void beta_skip_kernel(float* __restrict__ AGG, const float* __restrict__ SK,
    const float* __restrict__ Sm, const float* __restrict__ Wb)
{
  const int lane = threadIdx.x & 31;
  const int n = blockIdx.x * (blockDim.x >> 5) + (threadIdx.x >> 5);
  const float s0 = Sm[2 * (size_t)n]     + 1e-16f;
  const float s1 = Sm[2 * (size_t)n + 1] + 1e-16f;
  const float o0 = AGG[(size_t)n * HC + lane]      / s0;
  const float o1 = AGG[(size_t)n * HC + 32 + lane] / s1;
  const float x0 = SK [(size_t)n * HC + lane];
  const float x1 = SK [(size_t)n * HC + 32 + lane];
  float z = o0 * Wb[lane]      + x0 * Wb[64 + lane] + (o0 - x0) * Wb[128 + lane]
          + o1 * Wb[32 + lane] + x1 * Wb[96 + lane] + (o1 - x1) * Wb[160 + lane];
  z = wave_sum(z);
  const float beta = 1.f / (1.f + __expf(-z));
  AGG[(size_t)n * HC + lane]      = beta * x0 + (1.f - beta) * o0;
  AGG[(size_t)n * HC + 32 + lane] = beta * x1 + (1.f - beta) * o1;
}

// ---------------- batchnorm ----------------
__global__ void bn_stats_kernel(const float* __restrict__ H, float* __restrict__ stats) {
  int lane = threadIdx.x & 31;           // lane == column (EMB == 32)
  int wid  = (blockIdx.x * blockDim.x + threadIdx.x) >> 5;
  int nw   = (gridDim.x * blockDim.x) >> 5;
  float s = 0.f, q = 0.f;
  for (int r = wid; r < NN; r += nw) {
    float v = H[(size_t)r * EMB + lane];
    s += v; q += v * v;
  }
  atomicAdd(&stats[lane], s);
  atomicAdd(&stats[EMB + lane], q);
}

__global__ void bn_apply_kernel(float* __restrict__ H, const float* __restrict__ stats,
                                const float* __restrict__ g, const float* __restrict__ b) {
  int i = blockIdx.x * blockDim.x + threadIdx.x;
  if (i >= NN * EMB) return;
  int col = i & (EMB - 1);
  float mu  = stats[col] * (1.0f / NN);
  float var = stats[EMB + col] * (1.0f / NN) - mu * mu;
  H[i] = (H[i] - mu) * rsqrtf(var + BN_EPS) * g[col] + b[col];
}

// ---------------- host orchestration ----------------
extern "C" void kernel_launch(void* const* d_in, const int* in_sizes, int n_in,
                              void* d_out, int out_size, void* d_ws, size_t ws_size,
                              hipStream_t stream) {
  const float* x     = (const float*)d_in[0];
  const int*   ei    = (const int*)  d_in[1];
  const float* eattr = (const float*)d_in[2];
  const float* c1_Wq = (const float*)d_in[3];  const float* c1_bq = (const float*)d_in[4];
  const float* c1_Wk = (const float*)d_in[5];  const float* c1_bk = (const float*)d_in[6];
  const float* c1_Wv = (const float*)d_in[7];  const float* c1_bv = (const float*)d_in[8];
  const float* c1_We = (const float*)d_in[9];
  const float* c1_Ws = (const float*)d_in[10]; const float* c1_bs = (const float*)d_in[11];
  const float* c1_Wb = (const float*)d_in[12];
  const float* t1_W  = (const float*)d_in[13]; const float* t1_b  = (const float*)d_in[14];
  const float* bn1_g = (const float*)d_in[15]; const float* bn1_b = (const float*)d_in[16];
  const float* L_Wq  = (const float*)d_in[17]; const float* L_bq  = (const float*)d_in[18];
  const float* L_Wk  = (const float*)d_in[19]; const float* L_bk  = (const float*)d_in[20];
  const float* L_Wv  = (const float*)d_in[21]; const float* L_bv  = (const float*)d_in[22];
  const float* L_We  = (const float*)d_in[23];
  const float* L_Ws  = (const float*)d_in[24]; const float* L_bs  = (const float*)d_in[25];
  const float* L_Wb  = (const float*)d_in[26];
  const float* Lt_W  = (const float*)d_in[27]; const float* Lt_b  = (const float*)d_in[28];
  const float* Lbn_g = (const float*)d_in[29]; const float* Lbn_b = (const float*)d_in[30];
  const float* out_W = (const float*)d_in[31]; const float* out_b = (const float*)d_in[32];

  const int* srcp = ei;
  const int* dstp = ei + NE;

  // workspace layout (floats)
  float* ws = (float*)d_ws;
  size_t off = 0;
  float* Qb   = ws + off; off += (size_t)NN * HC;
  float* Kb   = ws + off; off += (size_t)NN * HC;
  float* Vb   = ws + off; off += (size_t)NN * HC;
  float* Sb   = ws + off; off += (size_t)NN * HC;   // skip projection
  float* AGG  = ws + off; off += (size_t)NN * HC;   // aggregated / conv output
  float* EP   = ws + off; off += (size_t)NE * HC;   // edge_attr @ We
  float* ALP  = ws + off; off += (size_t)NE * 2;
  float* Mx   = ws + off; off += (size_t)NN * 2;
  float* Sm   = ws + off; off += (size_t)NN * 2;
  float* H2a  = ws + off; off += (size_t)NN * EMB;
  float* H2b  = ws + off; off += (size_t)NN * EMB;
  float* stats= ws + off; off += 2 * EMB;
  (void)ws_size; (void)in_sizes; (void)n_in; (void)out_size;

  const int EWB = NE / 8;   // edge blocks, 8 waves (edges) per 256-thread block
  const int NWB = NN / 8;   // node blocks

  auto run_conv = [&](const float* hin, int Fin,
                      const float* Wq, const float* bq, const float* Wk, const float* bk,
                      const float* Wv, const float* bv, const float* We,
                      const float* Wsk, const float* bsk, const float* Wbeta) {
    // reset atomic accumulators
    fill_kernel<<<(NN * 2 + 255) / 256, 256, 0, stream>>>(Mx, -INFINITY, NN * 2);
    fill_kernel<<<(NN * 2 + 255) / 256, 256, 0, stream>>>(Sm, 0.f, NN * 2);
    fill_kernel<<<((size_t)NN * HC + 255) / 256, 256, 0, stream>>>(AGG, 0.f, NN * HC);
    // dense projections (WMMA, direct VMEM path)
    proj4_kernel<<<NN / 16, 128, 0, stream>>>(hin, Fin, Wq, bq, Wk, bk, Wv, bv,
                                              Wsk, bsk, Qb, Kb, Vb, Sb);
    // edge-attr projection (WMMA + TDM-staged B): [E,16] x [16,64]
    gemm_kernel<4, false><<<NE / 16, 32, (size_t)EDIM * HC * 4, stream>>>(
        eattr, EDIM, We, nullptr, EP);
    // attention (wave-per-edge, coalesced)
    edge_alpha_kernel<<<EWB, 256, 0, stream>>>(srcp, dstp, Qb, Kb, EP, ALP, Mx);
    edge_agg_kernel  <<<EWB, 256, 0, stream>>>(srcp, dstp, Vb, EP, ALP, Mx, Sm, AGG);
    // softmax-normalize + beta-gated skip (in place on AGG)
    beta_skip_kernel<<<NWB, 256, 0, stream>>>(AGG, Sb, Sm, Wbeta);
  };

  auto run_lin_bn = [&](const float* tW, const float* tb, const float* g, const float* b,
                        float* hout) {
    gemm_kernel<2, true><<<NN / 16, 32, (size_t)HC * EMB * 4, stream>>>(AGG, HC, tW, tb, hout);
    fill_kernel<<<1, 2 * EMB, 0, stream>>>(stats, 0.f, 2 * EMB);
    bn_stats_kernel<<<256, 256, 0, stream>>>(hout, stats);
    bn_apply_kernel<<<(NN * EMB + 255) / 256, 256, 0, stream>>>(hout, stats, g, b);
  };

  // ---- layer 1 ----
  run_conv(x, FEAT, c1_Wq, c1_bq, c1_Wk, c1_bk, c1_Wv, c1_bv, c1_We, c1_Ws, c1_bs, c1_Wb);
  run_lin_bn(t1_W, t1_b, bn1_g, bn1_b, H2a);

  // ---- inner layers ----
  float* hcur = H2a;
  float* hnext = H2b;
  for (int i = 0; i < NLAY; ++i) {
    run_conv(hcur, EMB,
             L_Wq + (size_t)i * EMB * HC, L_bq + (size_t)i * HC,
             L_Wk + (size_t)i * EMB * HC, L_bk + (size_t)i * HC,
             L_Wv + (size_t)i * EMB * HC, L_bv + (size_t)i * HC,
             L_We + (size_t)i * EDIM * HC,
             L_Ws + (size_t)i * EMB * HC, L_bs + (size_t)i * HC,
             L_Wb + (size_t)i * 3 * HC);
    run_lin_bn(Lt_W + (size_t)i * HC * EMB, Lt_b + (size_t)i * EMB,
               Lbn_g + (size_t)i * EMB, Lbn_b + (size_t)i * EMB, hnext);
    float* t = hcur; hcur = hnext; hnext = t;
  }

  // ---- classifier: [N,32] x [32,16] (WMMA + TDM) ----
  gemm_kernel<1, false><<<NN / 16, 32, (size_t)EMB * NCLS * 4, stream>>>(
      hcur, EMB, out_W, out_b, (float*)d_out);
}